// RadialBasisArbitraryLayerG_77386720740134
// MI455X (gfx1250) — compile-verified
//
#include <hip/hip_runtime.h>
#include <math.h>

typedef __attribute__((ext_vector_type(2))) float v2f;
typedef __attribute__((ext_vector_type(8))) float v8f;

#define BATCH 4
#define NPTS 512
#define IMG 512
#define C_FACTOR 2.0f
#define R_MAX_CAP 37.0f

// ---------------------------------------------------------------------------
// Kernel 0: zero the output image [B,2,H,W] (graph-capture-safe init)
// ---------------------------------------------------------------------------
__global__ void zero_out_kernel(float4* __restrict__ out, int n4) {
    int i = blockIdx.x * blockDim.x + threadIdx.x;
    if (i < n4) out[i] = make_float4(0.f, 0.f, 0.f, 0.f);
}

// ---------------------------------------------------------------------------
// Kernel 1: per-batch max over points of nearest-neighbor distance, * C_FACTOR.
// Gram matrix via V_WMMA_F32_16X16X4_F32:  dist^2 = n_i + n_j - 2 * (p_i . p_j)
// One workgroup (8 wave32) per batch. Points + norms staged in LDS.
// The row-tile index is pinned to an SGPR via readfirstlane (it is uniform
// within a wave by construction), so the diagonal-tile test is a scalar
// branch: the hot path has no exec-mask manipulation at all.
// ---------------------------------------------------------------------------
__global__ void nn_radius_kernel(const float* __restrict__ loc,
                                 float* __restrict__ r_out) {
    __shared__ float Px[NPTS];
    __shared__ float Py[NPTS];
    __shared__ float Nn[NPTS];
    __shared__ float wave_max_lds[8];

    const int b = blockIdx.x;
    const int tid = threadIdx.x;

    for (int i = tid; i < NPTS; i += blockDim.x) {
        float x = loc[(b * NPTS + i) * 2 + 0];
        float y = loc[(b * NPTS + i) * 2 + 1];
        Px[i] = x;
        Py[i] = y;
        Nn[i] = x * x + y * y;
    }
    __syncthreads();

    // wave id is uniform across the wave: make it an SGPR value
    const int wv   = __builtin_amdgcn_readfirstlane(tid >> 5);   // 0..7
    const int lane = tid & 31;
    const int l16  = lane & 15;
    const bool lo16 = (lane < 16);
    // D-element row offset within tile for this lane's half, and the v-index
    // that lands on the diagonal when ib == jt:
    const int half8 = lo16 ? 0 : 8;
    const int diag_v = l16 - half8;   // v == diag_v  <=>  row == col (diag tile)

    float wave_max = 0.0f;

    // 32 row-tiles of 16; wave wv owns tiles wv, wv+8, wv+16, wv+24
    for (int ib = wv; ib < NPTS / 16; ib += 8) {       // ib is SGPR-uniform
        // A tile (16x4): K=0,1 = (x,y) on lanes 0-15; K=2,3 zero on lanes 16-31
        const int arow = ib * 16 + l16;
        const float ax = Px[arow];
        const float ay = Py[arow];
        v2f a;
        a.x = lo16 ? ax : 0.0f;
        a.y = lo16 ? ay : 0.0f;

        // row norms for the 8 D VGPRs (fixed per i-block)
        float rn[8];
#pragma unroll
        for (int v = 0; v < 8; ++v) rn[v] = Nn[ib * 16 + v + half8];

        float m[8];
#pragma unroll
        for (int v = 0; v < 8; ++v) m[v] = 3.0e38f;

        for (int jt = 0; jt < NPTS / 16; ++jt) {
            const int bcol = jt * 16 + l16;
            const float bx = Px[bcol];
            const float by = Py[bcol];
            const float cn = Nn[bcol];     // col norm for this lane's D column
            v2f bb;
            bb.x = lo16 ? bx : 0.0f;
            bb.y = lo16 ? by : 0.0f;

            v8f g = {};
#if __has_builtin(__builtin_amdgcn_wmma_f32_16x16x4_f32)
            g = __builtin_amdgcn_wmma_f32_16x16x4_f32(
                    false, a, false, bb, (short)0, g, false, false);
#else
#pragma unroll
            for (int v = 0; v < 8; ++v) {
                int row = ib * 16 + v + half8;
                g[v] = Px[row] * Px[bcol] + Py[row] * Py[bcol];
            }
#endif
            if (ib != jt) {                // scalar compare -> s_cbranch
#pragma unroll
                for (int v = 0; v < 8; ++v) {
                    float d2 = fmaxf(fmaf(-2.0f, g[v], rn[v] + cn), 0.0f);
                    m[v] = fminf(m[v], d2);
                }
            } else {
#pragma unroll
                for (int v = 0; v < 8; ++v) {
                    float d2 = fmaxf(fmaf(-2.0f, g[v], rn[v] + cn), 0.0f);
                    d2 = (v == diag_v) ? 3.0e38f : d2;   // mask diagonal
                    m[v] = fminf(m[v], d2);
                }
            }
        }

        // min over the 16 columns inside each 16-lane half (xor masks stay in-half)
#pragma unroll
        for (int v = 0; v < 8; ++v) {
            float x = m[v];
            x = fminf(x, __shfl_xor(x, 1, 32));
            x = fminf(x, __shfl_xor(x, 2, 32));
            x = fminf(x, __shfl_xor(x, 4, 32));
            x = fminf(x, __shfl_xor(x, 8, 32));
            wave_max = fmaxf(wave_max, sqrtf(x));   // max over rows
        }
    }

    // max across all 32 lanes of the wave
    wave_max = fmaxf(wave_max, __shfl_xor(wave_max, 16, 32));
    wave_max = fmaxf(wave_max, __shfl_xor(wave_max, 8, 32));
    wave_max = fmaxf(wave_max, __shfl_xor(wave_max, 4, 32));
    wave_max = fmaxf(wave_max, __shfl_xor(wave_max, 2, 32));
    wave_max = fmaxf(wave_max, __shfl_xor(wave_max, 1, 32));
    if (lane == 0) wave_max_lds[wv] = wave_max;
    __syncthreads();

    if (tid == 0) {
        float mx = 0.0f;
#pragma unroll
        for (int i = 0; i < 8; ++i) mx = fmaxf(mx, wave_max_lds[i]);
        r_out[b] = mx * C_FACTOR;
    }
}

// ---------------------------------------------------------------------------
// Kernel 2: r_max = min(ceil(max_b r[b]), 37)  -- single thread
// ---------------------------------------------------------------------------
__global__ void rmax_kernel(const float* __restrict__ r_b,
                            int* __restrict__ rmax_out) {
    float mx = 0.0f;
#pragma unroll
    for (int i = 0; i < BATCH; ++i) mx = fmaxf(mx, r_b[i]);
    float c = fminf(ceilf(mx), R_MAX_CAP);
    rmax_out[0] = (int)c;
}

// ---------------------------------------------------------------------------
// Kernel 3: Wendland-C2 splat. One block per control point; 16x16 thread tile
// strided over the (2*r_max)^2 window -- no runtime integer division, y-row
// math hoisted out of the inner loop. Scatter-add into [B,2,H,W] via L2 f32
// atomics (8 MB output is fully L2-resident on MI455X).
// ---------------------------------------------------------------------------
__global__ void splat_kernel(const float* __restrict__ loc,
                             const float* __restrict__ alpha,
                             const float* __restrict__ r_b,
                             const int* __restrict__ rmax_p,
                             float* __restrict__ out) {
    const int pt = blockIdx.x;          // 0 .. B*N-1
    const int b  = pt / NPTS;           // power-of-two divide -> shift

    const int r_max = rmax_p[0];
    const int w2    = 2 * r_max;

    const float px = loc[pt * 2 + 0];
    const float py = loc[pt * 2 + 1];
    const float lo = (float)r_max;
    const float hi = (float)IMG - (float)r_max;   // both coords clamp with I_SIZE[1]
    const float txc = fminf(fmaxf(px, lo), hi);
    const float tyc = fminf(fmaxf(py, lo), hi);

    const float rinv = 1.0f / r_b[b];
    const float a0 = alpha[pt * 2 + 0];
    const float a1 = alpha[pt * 2 + 1];

    float* __restrict__ out0 = out + ((size_t)(b * 2 + 0)) * IMG * IMG;
    float* __restrict__ out1 = out + ((size_t)(b * 2 + 1)) * IMG * IMG;

    const int tx = threadIdx.x & 15;
    const int ty = threadIdx.x >> 4;

    for (int iy = ty; iy < w2; iy += 16) {
        const float gy = (float)(iy - r_max);
        const float fy = floorf(gy + tyc);
        const float dy = fy - py;
        const float dy2 = dy * dy;
        const int yoff = (int)fy * IMG;           // clamp guarantees in-bounds
        for (int ix = tx; ix < w2; ix += 16) {
            const float gx = (float)(ix - r_max);
            const float fx = floorf(gx + txc);
            const float dx = fx - px;
            const float dist = sqrtf(fmaf(dx, dx, dy2)) * rinv;
            if (dist < 1.0f) {
                const float om  = 1.0f - dist;
                const float om2 = om * om;
                const float wgt = om2 * om2 * fmaf(4.0f, dist, 1.0f);
                const int o = yoff + (int)fx;
                atomicAdd(&out0[o], wgt * a0);
                atomicAdd(&out1[o], wgt * a1);
            }
        }
    }
}

// ---------------------------------------------------------------------------
extern "C" void kernel_launch(void* const* d_in, const int* in_sizes, int n_in,
                              void* d_out, int out_size, void* d_ws, size_t ws_size,
                              hipStream_t stream) {
    const float* loc   = (const float*)d_in[0];   // [B, N, 2] (x, y)
    const float* alpha = (const float*)d_in[1];   // [B, N, 2]
    float* out = (float*)d_out;                   // [B, 2, H, W]

    float* rb    = (float*)d_ws;                  // [B] per-batch radius
    int*   rmaxp = (int*)((float*)d_ws + BATCH);  // scalar window half-size

    // zero the output image (deterministic init each launch)
    const int n4 = out_size / 4;
    zero_out_kernel<<<(n4 + 255) / 256, 256, 0, stream>>>((float4*)out, n4);

    // per-batch radius via WMMA Gram matrix
    nn_radius_kernel<<<BATCH, 256, 0, stream>>>(loc, rb);

    // scalar r_max
    rmax_kernel<<<1, 1, 0, stream>>>(rb, rmaxp);

    // splat
    splat_kernel<<<BATCH * NPTS, 256, 0, stream>>>(loc, alpha, rb, rmaxp, out);
}